// Model_11888469475940
// MI455X (gfx1250) — compile-verified
//
#include <hip/hip_runtime.h>

// Problem constants (match reference)
#define BB 8
#define CC 32
#define DD 64
#define HH 64
#define WW 64
#define LFLAT 32768     // 32*32*32 pooled elements per (b,c)
#define TSEQ 16
#define AVGK 2048       // LFLAT / TSEQ
#define HID 128
#define EPSF 1e-5f

typedef __attribute__((ext_vector_type(2))) float v2f;
typedef __attribute__((ext_vector_type(4))) float v4f;
typedef __attribute__((ext_vector_type(8))) float v8f;

// ---------------------------------------------------------------------------
// Kernel 1: fused MaxPool3d(2)+argmax, segment sums (AvgPool1d seq) and
// sum/sumsq partials (BatchNorm batch stats).
// grid = B*C*16 blocks (one per (b,c,t) segment of 2048 pooled elems),
// 256 threads, TWO adjacent W-windows per thread per iteration so the
// streaming x reads are 16-byte global_load_b128 with NT hint (keep L2 for
// the pooled/amax intermediate, which IS L2-resident: 40 MB << 192 MB).
// ---------------------------------------------------------------------------
__global__ __launch_bounds__(256) void pool_kernel(
    const float* __restrict__ x,
    float* __restrict__ pooled,
    unsigned short* __restrict__ amax2,  // packed 2 window-indices per u16
    float* __restrict__ seg_sum,         // [B*C*16] deterministic block sums
    float* __restrict__ seg_sumsq)       // [B*C*16]
{
    const int blk = blockIdx.x;          // b*512 + c*16 + t
    const int t   = blk & 15;
    const int c   = (blk >> 4) & 31;
    const int b   = blk >> 9;
    const int tid = threadIdx.x;

    const float* xin = x + (size_t)(b * CC + c) * (DD * HH * WW);
    float* pout      = pooled + (size_t)(b * CC + c) * LFLAT;
    unsigned short* ao = amax2 + ((size_t)(b * CC + c) * LFLAT >> 1);

    float lsum = 0.f, lsq = 0.f;
    #pragma unroll
    for (int i = 0; i < 4; ++i) {
        const int flat = t * AVGK + i * 512 + tid * 2;   // even
        const int wp = flat & 31;                         // even
        const int hp = (flat >> 5) & 31;
        const int dp = flat >> 10;
        const float* base = xin + ((size_t)(dp * 2) * HH + hp * 2) * WW + wp * 2;
        // four rows of 4 floats cover two adjacent 2x2x2 windows (NT stream)
        const v4f r00 = __builtin_nontemporal_load((const v4f*)(base));
        const v4f r01 = __builtin_nontemporal_load((const v4f*)(base + WW));
        const v4f r10 = __builtin_nontemporal_load((const v4f*)(base + HH * WW));
        const v4f r11 = __builtin_nontemporal_load((const v4f*)(base + HH * WW + WW));
        // window-local index = kd*4 + kh*2 + kw, first-max wins (argmax)
        float v0[8] = { r00.x, r00.y, r01.x, r01.y, r10.x, r10.y, r11.x, r11.y };
        float v1[8] = { r00.z, r00.w, r01.z, r01.w, r10.z, r10.w, r11.z, r11.w };
        float best0 = v0[0], best1 = v1[0];
        int bi0 = 0, bi1 = 0;
        #pragma unroll
        for (int j = 1; j < 8; ++j) {
            if (v0[j] > best0) { best0 = v0[j]; bi0 = j; }
            if (v1[j] > best1) { best1 = v1[j]; bi1 = j; }
        }
        v2f pv = { best0, best1 };
        *(v2f*)(pout + flat) = pv;                       // RT: stays in L2
        ao[flat >> 1] = (unsigned short)(bi0 | (bi1 << 8));
        lsum += best0 + best1;
        lsq  += best0 * best0 + best1 * best1;
    }

    __shared__ float ssum[256];
    __shared__ float ssq[256];
    ssum[tid] = lsum;
    ssq[tid]  = lsq;
    __syncthreads();
    #pragma unroll
    for (int s = 128; s > 0; s >>= 1) {
        if (tid < s) { ssum[tid] += ssum[tid + s]; ssq[tid] += ssq[tid + s]; }
        __syncthreads();
    }
    if (tid == 0) {
        seg_sum[blk]   = ssum[0];
        seg_sumsq[blk] = ssq[0];
    }
}

// ---------------------------------------------------------------------------
// Kernel 2: single workgroup (256 threads = 8 wave32s). Does:
//  - seq[t][b][c] and pooled_mean from segment sums
//  - per-channel BN mean / inv-std (deterministic reduction)
//  - h0 = tanh(Xmean @ W_init^T + b_init) via v_wmma_f32_16x16x4_f32
//  - 16 RNN steps: h = tanh(seq_t @ W_ih^T + h @ W_hh^T + b_ih + b_hh)
//  - mod = sigmoid(h @ W_mod^T + b_mod); gain = mod + 0.5
//  - per-(b,c) affine coefficients: out = A*pooled + B
// Batch 8 padded to a 16-row M tile; wave w owns output columns [16w,16w+16).
// Real batch rows 0..7 land in lanes 0..15 of each wave (D-layout VGPR 0..7).
// ---------------------------------------------------------------------------
__global__ __launch_bounds__(256) void rnn_mod_kernel(
    const float* __restrict__ seg_sum,
    const float* __restrict__ seg_sumsq,
    const float* __restrict__ W_init, const float* __restrict__ b_init,
    const float* __restrict__ W_ih,   const float* __restrict__ b_ih,
    const float* __restrict__ W_hh,   const float* __restrict__ b_hh,
    const float* __restrict__ W_mod,  const float* __restrict__ b_mod,
    const float* __restrict__ gamma,  const float* __restrict__ beta,
    float* __restrict__ A_coef,       // [B*C]
    float* __restrict__ B_coef)       // [B*C]
{
    __shared__ float X0[16][CC];            // pooled_mean, rows 8..15 zero
    __shared__ float SEQ[TSEQ][16][CC];     // seq inputs, rows 8..15 zero
    __shared__ float Hs[16][HID];           // hidden state
    __shared__ float GainS[BB][CC];
    __shared__ float chMean[CC];
    __shared__ float chInvStd[CC];

    const int tid = threadIdx.x;

    // stage sequence inputs: seq = seg_sum / 2048
    for (int i = tid; i < TSEQ * 16 * CC; i += 256) {
        const int t = i / (16 * CC);
        const int r = (i / CC) & 15;
        const int c = i % CC;
        float v = 0.f;
        if (r < BB) v = seg_sum[(r * CC + c) * TSEQ + t] * (1.0f / (float)AVGK);
        SEQ[t][r][c] = v;
    }
    // pooled_mean (h0 input)
    for (int i = tid; i < 16 * CC; i += 256) {
        const int r = i / CC;
        const int c = i % CC;
        float v = 0.f;
        if (r < BB) {
            float s = 0.f;
            for (int t = 0; t < TSEQ; ++t) s += seg_sum[(r * CC + c) * TSEQ + t];
            v = s * (1.0f / (float)LFLAT);
        }
        X0[r][c] = v;
    }
    // per-channel BN stats (deterministic, fixed order)
    if (tid < CC) {
        float s = 0.f, q = 0.f;
        for (int b = 0; b < BB; ++b)
            for (int t = 0; t < TSEQ; ++t) {
                const int idx = b * (CC * TSEQ) + tid * TSEQ + t;
                s += seg_sum[idx];
                q += seg_sumsq[idx];
            }
        const float invN = 1.0f / ((float)BB * (float)LFLAT);
        const float m = s * invN;
        chMean[tid]   = m;
        chInvStd[tid] = rsqrtf(q * invN - m * m + EPSF);
    }
    __syncthreads();

    const int lane = tid & 31;
    const int wave = tid >> 5;      // output N tile 0..7
    const int halh = lane >> 4;     // 0/1: which K pair / which D row half
    const int l16  = lane & 15;
    const int n    = wave * 16 + l16;   // output column 0..127

    // ---- h0 = tanh(X0 @ W_init^T + b_init), K = 32 ----
    v8f acc = {};
    for (int kk = 0; kk < 8; ++kk) {
        const int k0 = kk * 4 + 2 * halh;
        v2f a  = { X0[l16][k0], X0[l16][k0 + 1] };
        v2f bb = { W_init[n * CC + k0], W_init[n * CC + k0 + 1] };
        acc = __builtin_amdgcn_wmma_f32_16x16x4_f32(false, a, false, bb,
                                                    (short)0, acc, false, false);
    }
    {
        const float bi = b_init[n];
        #pragma unroll
        for (int r = 0; r < 8; ++r)
            Hs[r + 8 * halh][n] = tanhf(acc[r] + bi);
    }
    __syncthreads();

    // ---- RNN scan: 16 steps ----
    for (int t = 0; t < TSEQ; ++t) {
        v8f a2 = {};
        for (int kk = 0; kk < 8; ++kk) {       // seq_t @ W_ih^T, K=32
            const int k0 = kk * 4 + 2 * halh;
            v2f a  = { SEQ[t][l16][k0], SEQ[t][l16][k0 + 1] };
            v2f bb = { W_ih[n * CC + k0], W_ih[n * CC + k0 + 1] };
            a2 = __builtin_amdgcn_wmma_f32_16x16x4_f32(false, a, false, bb,
                                                       (short)0, a2, false, false);
        }
        for (int kk = 0; kk < 32; ++kk) {      // h @ W_hh^T, K=128
            const int k0 = kk * 4 + 2 * halh;
            v2f a  = { Hs[l16][k0], Hs[l16][k0 + 1] };
            v2f bb = { W_hh[n * HID + k0], W_hh[n * HID + k0 + 1] };
            a2 = __builtin_amdgcn_wmma_f32_16x16x4_f32(false, a, false, bb,
                                                       (short)0, a2, false, false);
        }
        const float bias = b_ih[n] + b_hh[n];
        float nh[8];
        #pragma unroll
        for (int r = 0; r < 8; ++r) nh[r] = tanhf(a2[r] + bias);
        __syncthreads();                        // all reads of old h done
        #pragma unroll
        for (int r = 0; r < 8; ++r) Hs[r + 8 * halh][n] = nh[r];
        __syncthreads();
    }

    // ---- mod = sigmoid(h @ W_mod^T + b_mod), output N=32 -> waves 0,1 ----
    if (wave < 2) {
        v8f a3 = {};
        for (int kk = 0; kk < 32; ++kk) {
            const int k0 = kk * 4 + 2 * halh;
            v2f a  = { Hs[l16][k0], Hs[l16][k0 + 1] };
            v2f bb = { W_mod[n * HID + k0], W_mod[n * HID + k0 + 1] };
            a3 = __builtin_amdgcn_wmma_f32_16x16x4_f32(false, a, false, bb,
                                                       (short)0, a3, false, false);
        }
        if (halh == 0) {   // rows 0..7 = real batch
            const float bm = b_mod[n];
            #pragma unroll
            for (int r = 0; r < 8; ++r) {
                const float m = 1.0f / (1.0f + __expf(-(a3[r] + bm)));
                GainS[r][n] = m + 0.5f;      // 1 + (mod - 0.5)
            }
        }
    }
    __syncthreads();

    // ---- fold BN + modulation into per-(b,c) affine: out = A*pooled + B ----
    {
        const int b = tid >> 5;
        const int c = tid & 31;
        const float inv = chInvStd[c];
        const float mn  = chMean[c];
        const float g   = GainS[b][c];
        A_coef[tid] = gamma[c] * inv * g;
        B_coef[tid] = (beta[c] - gamma[c] * inv * mn) * g;
    }
}

// ---------------------------------------------------------------------------
// Kernel 3: MaxUnpool3d scatter. Two adjacent windows per thread so every
// output row write is a 16-byte global_store_b128 with NT hint (out is
// write-once streaming; pooled/amax come hot out of L2).
// grid = B*C*64 blocks of 256 threads (512 pooled elems per block).
// ---------------------------------------------------------------------------
__global__ __launch_bounds__(256) void unpool_kernel(
    const float* __restrict__ pooled,
    const unsigned short* __restrict__ amax2,
    const float* __restrict__ A_coef,
    const float* __restrict__ B_coef,
    float* __restrict__ out)
{
    const int blk   = blockIdx.x;
    const int bc    = blk >> 6;       // b*32 + c
    const int chunk = blk & 63;
    const int tid   = threadIdx.x;
    const int flat  = chunk * 512 + tid * 2;   // even
    const int wp = flat & 31;                   // even
    const int hp = (flat >> 5) & 31;
    const int dp = flat >> 10;

    const v2f p2 = *(const v2f*)(pooled + (size_t)bc * LFLAT + flat);
    const unsigned short aa = amax2[((size_t)bc * LFLAT + flat) >> 1];
    const float Ac = A_coef[bc], Bc = B_coef[bc];
    const float pm0 = Ac * p2.x + Bc;
    const float pm1 = Ac * p2.y + Bc;
    const int a0 = aa & 0xff;
    const int a1 = aa >> 8;

    float* obase = out + (size_t)bc * (DD * HH * WW)
                       + ((size_t)(dp * 2) * HH + hp * 2) * WW + wp * 2;
    #pragma unroll
    for (int kd = 0; kd < 2; ++kd) {
        #pragma unroll
        for (int kh = 0; kh < 2; ++kh) {
            const int j0 = kd * 4 + kh * 2;
            v4f w;
            w.x = (a0 == j0)     ? pm0 : 0.0f;
            w.y = (a0 == j0 + 1) ? pm0 : 0.0f;
            w.z = (a1 == j0)     ? pm1 : 0.0f;
            w.w = (a1 == j0 + 1) ? pm1 : 0.0f;
            __builtin_nontemporal_store(w, (v4f*)(obase + (size_t)kd * HH * WW + kh * WW));
        }
    }
}

// ---------------------------------------------------------------------------
extern "C" void kernel_launch(void* const* d_in, const int* in_sizes, int n_in,
                              void* d_out, int out_size, void* d_ws, size_t ws_size,
                              hipStream_t stream) {
    const float* x      = (const float*)d_in[0];
    const float* W_init = (const float*)d_in[1];
    const float* b_init = (const float*)d_in[2];
    const float* W_ih   = (const float*)d_in[3];
    const float* b_ih   = (const float*)d_in[4];
    const float* W_hh   = (const float*)d_in[5];
    const float* b_hh   = (const float*)d_in[6];
    const float* W_mod  = (const float*)d_in[7];
    const float* b_mod  = (const float*)d_in[8];
    const float* gamma  = (const float*)d_in[9];
    const float* beta   = (const float*)d_in[10];
    float* out = (float*)d_out;

    // workspace layout
    char* ws = (char*)d_ws;
    const size_t pooled_bytes = (size_t)BB * CC * LFLAT * sizeof(float);   // 32 MB
    const size_t amax_bytes   = (size_t)BB * CC * LFLAT;                    // 8 MB
    const size_t seg_bytes    = (size_t)BB * CC * TSEQ * sizeof(float);     // 16 KB
    float*          pooled    = (float*)ws;
    unsigned short* amax2     = (unsigned short*)(ws + pooled_bytes);
    float*          seg_sum   = (float*)(ws + pooled_bytes + amax_bytes);
    float*          seg_sumsq = (float*)(ws + pooled_bytes + amax_bytes + seg_bytes);
    float*          A_coef    = (float*)(ws + pooled_bytes + amax_bytes + 2 * seg_bytes);
    float*          B_coef    = A_coef + BB * CC;

    pool_kernel<<<BB * CC * TSEQ, 256, 0, stream>>>(x, pooled, amax2, seg_sum, seg_sumsq);
    rnn_mod_kernel<<<1, 256, 0, stream>>>(seg_sum, seg_sumsq,
                                          W_init, b_init, W_ih, b_ih, W_hh, b_hh,
                                          W_mod, b_mod, gamma, beta, A_coef, B_coef);
    unpool_kernel<<<BB * CC * (LFLAT / 512), 256, 0, stream>>>(pooled, amax2, A_coef, B_coef, out);
}